// VIN_83124797047494
// MI455X (gfx1250) — compile-verified
//
#include <hip/hip_runtime.h>

typedef float v2f __attribute__((ext_vector_type(2)));
typedef float v8f __attribute__((ext_vector_type(8)));

#define HH 64
#define PS 66            // padded row stride (64 + 2 halo)
#define PA (PS*PS)       // padded area = 4356 floats

// ---------------------------------------------------------------------------
// Pre-kernel: collapse hidden layer (conv+bias then 1x1 conv are both linear).
// W_eff[ci,ky,kx] = sum_h W_r[h] * W_h[h,ci,ky,kx]   (18 values)
// b_eff          = sum_h W_r[h] * b_h[h]             (w_eff[18])
// ---------------------------------------------------------------------------
__global__ void vin_weff_kernel(const float* __restrict__ W_h,
                                const float* __restrict__ b_h,
                                const float* __restrict__ W_r,
                                float* __restrict__ w_eff)
{
    int t = threadIdx.x;
    if (t < 18) {
        float s = 0.0f;
        for (int h = 0; h < 150; ++h) s += W_r[h] * W_h[h * 18 + t];
        w_eff[t] = s;
    } else if (t == 18) {
        float s = 0.0f;
        for (int h = 0; h < 150; ++h) s += W_r[h] * b_h[h];
        w_eff[18] = s;
    }
}

// ---------------------------------------------------------------------------
// Main kernel: one workgroup (256 threads = 8 wave32) per image.
// LDS: reward (padded) + double-buffered v (padded), all LDS-resident.
// VI conv as GEMM per 16-pixel tile via V_WMMA_F32_16X16X4_F32:
//   M=16 (10 out-channels padded), N=16 pixels, K=20 (2ch x 9 taps, padded)
//   -> 5 K-chunks; chunk 4 holds k=16..19 (k>=18 zeroed on lanes 16-31).
// ---------------------------------------------------------------------------
__global__ void __launch_bounds__(256)
vin_vi_kernel(const float* __restrict__ input,
              const int*   __restrict__ state_x,
              const int*   __restrict__ state_y,
              const int*   __restrict__ num_vi_p,
              const float* __restrict__ W_q,
              const float* __restrict__ W_v,
              const float* __restrict__ W_fc,
              const float* __restrict__ w_eff,
              float*       __restrict__ out,
              int B)
{
    // [0,PA): reward   [PA,2PA): v buffer 0   [2PA,3PA): v buffer 1
    __shared__ float smem[3 * PA];
    __shared__ float wqv[10 * 18];     // W_qv[c][k], k = ch*9 + ky*3 + kx
    __shared__ float qxy[10];
    __shared__ float logit[8];

    const int b    = blockIdx.x;
    const int tid  = threadIdx.x;
    const int lane = tid & 31;
    const int wave = tid >> 5;

    // Zero reward halo + both v buffers (halo stays zero forever).
    for (int i = tid; i < 3 * PA; i += 256) smem[i] = 0.0f;
    if (tid < 180) {
        int c = tid / 18, k = tid - c * 18;
        wqv[tid] = (k < 9) ? W_q[c * 9 + k] : W_v[c * 9 + (k - 9)];
    }
    __syncthreads();

    // ---- reward = conv3x3(input, W_eff) + b_eff (collapsed hidden layer) ----
    {
        float we[18];
#pragma unroll
        for (int i = 0; i < 18; ++i) we[i] = w_eff[i];
        const float beff = w_eff[18];
        const float* inb = input + (size_t)b * 2 * HH * HH;
        for (int p = tid; p < HH * HH; p += 256) {
            int r = p >> 6, c = p & 63;
            float acc = beff;
#pragma unroll
            for (int ci = 0; ci < 2; ++ci) {
#pragma unroll
                for (int ky = 0; ky < 3; ++ky) {
                    int rr = r + ky - 1;
                    if (rr < 0 || rr >= HH) continue;
#pragma unroll
                    for (int kx = 0; kx < 3; ++kx) {
                        int cc = c + kx - 1;
                        if (cc < 0 || cc >= HH) continue;
                        acc += we[ci * 9 + ky * 3 + kx] * inb[ci * HH * HH + rr * HH + cc];
                    }
                }
            }
            smem[(r + 1) * PS + (c + 1)] = acc;
        }
    }
    __syncthreads();

    // ---- Per-lane A-matrix registers and B-gather descriptors ----
    // 16x4 f32 A layout: lanes 0-15 -> M=lane, VGPR0=K0, VGPR1=K1;
    //                    lanes 16-31 -> M=lane-16, VGPR0=K2, VGPR1=K3.
    // B (4x16) mirrors:  lanes 0-15 -> N=lane, K0/K1; lanes 16-31 -> N, K2/K3.
    const int mrow = lane & 15;
    const int kb   = (lane >> 4) << 1;   // 0 or 2
    const bool lo  = (lane < 16);

    v2f A[5];
#pragma unroll
    for (int kc = 0; kc < 5; ++kc) {
        int k0 = 4 * kc + kb;
        float a0 = (k0 < 18 && mrow < 10)     ? wqv[mrow * 18 + k0]     : 0.0f;
        float a1 = (k0 + 1 < 18 && mrow < 10) ? wqv[mrow * 18 + k0 + 1] : 0.0f;
        A[kc].x = a0;
        A[kc].y = a1;
    }

    // Chunks 0-3 cover k = 4kc + kb + j <= 15 < 18 for every lane: always a
    // real patch element (reward if k<9, v otherwise) -> no masking needed.
    int dlt[8];
    int isv[8];
#pragma unroll
    for (int kc = 0; kc < 4; ++kc) {
#pragma unroll
        for (int j = 0; j < 2; ++j) {
            int k = 4 * kc + kb + j;
            int i = kc * 2 + j;
            if (k < 9) { isv[i] = 0; dlt[i] = (k / 3) * PS + (k % 3); }
            else       { int kk = k - 9; isv[i] = 1; dlt[i] = (kk / 3) * PS + (kk % 3); }
        }
    }
    // Chunk 4: lanes 0-15 -> k=16,17 (v-patch kk=7,8 -> deltas 2*PS+1, 2*PS+2);
    //          lanes 16-31 -> k=18,19 (K padding) -> zeroed via cndmask.
    const int d4_0 = 2 * PS + 1;
    const int d4_1 = 2 * PS + 2;

    // ---- Value-iteration loop: v <- max_c conv3x3([reward, v], W_qv) ----
    const int NVI = num_vi_p[0];   // 36
    int cur = 0;
    for (int it = 0; it < NVI; ++it) {
        const int voff_cur = PA * (1 + cur);
        const int voff_nxt = PA * (2 - cur);

        // Hoist base selection out of the tile loop (iteration-invariant).
        int aoff[8];
#pragma unroll
        for (int i = 0; i < 8; ++i) aoff[i] = dlt[i] + (isv[i] ? voff_cur : 0);
        const int a4_0 = voff_cur + d4_0;
        const int a4_1 = voff_cur + d4_1;

        // 256 tiles/image: 64 rows x 4 segments of 16 pixels; 8 waves.
        for (int t = wave; t < 256; t += 8) {
            const int row = t >> 2;
            const int col = ((t & 3) << 4) + (lane & 15);
            const int pix = row * PS + col;   // top-left of 3x3 window (padded)
            v8f acc = {};
#pragma unroll
            for (int kc = 0; kc < 4; ++kc) {
                v2f Bv;
                Bv.x = smem[pix + aoff[kc * 2]];
                Bv.y = smem[pix + aoff[kc * 2 + 1]];
                acc = __builtin_amdgcn_wmma_f32_16x16x4_f32(
                          false, A[kc], false, Bv, (short)0, acc, false, false);
            }
            {
                float b0 = smem[pix + a4_0];
                float b1 = smem[pix + a4_1];
                v2f B4;
                B4.x = lo ? b0 : 0.0f;
                B4.y = lo ? b1 : 0.0f;
                acc = __builtin_amdgcn_wmma_f32_16x16x4_f32(
                          false, A[4], false, B4, (short)0, acc, false, false);
            }
            // Channel max: lanes 0-15 hold M=0..7; lanes 16-31 hold M=8,9 in
            // D VGPR0/1 (padded channels 10-15 never enter the max).
            float m07 = fmaxf(fmaxf(fmaxf(acc[0], acc[1]), fmaxf(acc[2], acc[3])),
                              fmaxf(fmaxf(acc[4], acc[5]), fmaxf(acc[6], acc[7])));
            float m89 = fmaxf(acc[0], acc[1]);
            float mv  = lo ? m07 : m89;
            mv = fmaxf(mv, __shfl_xor(mv, 16, 32));
            if (lo) smem[voff_nxt + pix + PS + 1] = mv;
        }
        __syncthreads();
        cur ^= 1;
    }

    // ---- Final q at (state_x, state_y) only, then FC + softmax ----
    const int voff_fin = PA * (1 + cur);
    if (tid < 10) {
        const int r = state_x[b], c = state_y[b];
        const int pb = r * PS + c;
        float q = 0.0f;
#pragma unroll
        for (int k = 0; k < 9; ++k) {
            q += wqv[tid * 18 + k]     * smem[pb + (k / 3) * PS + (k % 3)];
            q += wqv[tid * 18 + 9 + k] * smem[voff_fin + pb + (k / 3) * PS + (k % 3)];
        }
        qxy[tid] = q;
    }
    __syncthreads();
    if (tid < 8) {
        float l = 0.0f;
#pragma unroll
        for (int c = 0; c < 10; ++c) l += W_fc[tid * 10 + c] * qxy[c];
        logit[tid] = l;
        out[(size_t)b * 8 + tid] = l;
    }
    __syncthreads();
    if (tid < 8) {
        float mx = logit[0];
#pragma unroll
        for (int j = 1; j < 8; ++j) mx = fmaxf(mx, logit[j]);
        float s = 0.0f;
#pragma unroll
        for (int j = 0; j < 8; ++j) s += expf(logit[j] - mx);
        out[(size_t)B * 8 + (size_t)b * 8 + tid] = expf(logit[tid] - mx) / s;
    }
}

extern "C" void kernel_launch(void* const* d_in, const int* in_sizes, int n_in,
                              void* d_out, int out_size, void* d_ws, size_t ws_size,
                              hipStream_t stream)
{
    const float* input  = (const float*)d_in[0];   // (B,2,64,64)
    const int*   sx     = (const int*)  d_in[1];   // (B,)
    const int*   sy     = (const int*)  d_in[2];   // (B,)
    const int*   nvi    = (const int*)  d_in[3];   // scalar 36
    const float* W_h    = (const float*)d_in[4];   // (150,2,3,3)
    const float* b_h    = (const float*)d_in[5];   // (150,)
    const float* W_r    = (const float*)d_in[6];   // (1,150,1,1)
    const float* W_q    = (const float*)d_in[7];   // (10,1,3,3)
    const float* W_v    = (const float*)d_in[8];   // (10,1,3,3)
    const float* W_fc   = (const float*)d_in[9];   // (8,10)
    float*       out    = (float*)d_out;           // logits(B,8) ++ softmax(B,8)
    float*       w_eff  = (float*)d_ws;            // 19 floats scratch

    const int B = in_sizes[1];

    vin_weff_kernel<<<1, 32, 0, stream>>>(W_h, b_h, W_r, w_eff);
    vin_vi_kernel<<<B, 256, 0, stream>>>(input, sx, sy, nvi,
                                         W_q, W_v, W_fc, w_eff, out, B);
}